// AttributeEmbeddingLayer_31653908972243
// MI455X (gfx1250) — compile-verified
//
#include <hip/hip_runtime.h>

// MI455X / gfx1250 implementation.
// Sizes fixed by the reference: N=8192 per side, E=128, D=64, K=32, M=4.

typedef float v2f __attribute__((ext_vector_type(2)));
typedef float v8f __attribute__((ext_vector_type(8)));

constexpr int N = 8192;
constexpr int E = 128;
constexpr int D = 64;
constexpr int K = 32;
constexpr int M = 4;

// Native CDNA5 V_TANH_F32 if the toolchain exposes it; ocml fallback otherwise.
__device__ __forceinline__ float fast_tanh(float x) {
#if __has_builtin(__builtin_amdgcn_tanhf)
  return __builtin_amdgcn_tanhf(x);
#elif __has_builtin(__builtin_amdgcn_tanh_f32)
  return __builtin_amdgcn_tanh_f32(x);
#else
  return tanhf(x);
#endif
}

// ---------------------------------------------------------------------------
// P[m, i, d] = sum_e Emb[i, e] * W[m, e, d]     (W: [M, E, D], P: [M, N, D])
// One wave per 16x16 output tile, fp32 WMMA 16x16x4, 32 MACs over K=128.
// Fragment layout per CDNA5 ISA 7.12.2:
//   A (16x4):  lane L row=L&15; vgpr j holds K = j + 2*(L>>4)
//   B (4x16):  lane L col=L&15; vgpr j holds K = j + 2*(L>>4)
//   C (16x16): vgpr r -> row = r + 8*(L>>4), col = L&15
// ---------------------------------------------------------------------------
__global__ __launch_bounds__(256) void gemm_proj_kernel(
    const float* __restrict__ Emb, const float* __restrict__ W,
    float* __restrict__ P)
{
  const int lane = threadIdx.x & 31;
  const int wid  = threadIdx.x >> 5;
  const int tile = blockIdx.x * 8 + wid;     // 0 .. M*(N/16)*(D/16)-1 = 8191
  const int m    = tile >> 11;               // / ((N/16)*(D/16)) = /2048
  const int rem  = tile & 2047;
  const int rt   = rem >> 2;                 // row tile 0..511
  const int dt   = rem & 3;                  // col tile 0..3
  const int row  = lane & 15;
  const int hi   = lane >> 4;

  const float* a_row = Emb + (size_t)(rt * 16 + row) * E + 2 * hi;
  const float* b_col = W + (size_t)m * E * D + (size_t)(2 * hi) * D + dt * 16 + row;

  v8f c = {};
#pragma unroll 4
  for (int k = 0; k < E; k += 4) {
    v2f a, b;
    a[0] = a_row[k];
    a[1] = a_row[k + 1];
    b[0] = b_col[(size_t)k * D];
    b[1] = b_col[(size_t)(k + 1) * D];
    c = __builtin_amdgcn_wmma_f32_16x16x4_f32(false, a, false, b,
                                              (short)0, c, false, false);
  }

  float* out = P + (size_t)m * N * D + (size_t)(rt * 16 + 8 * hi) * D + dt * 16 + row;
#pragma unroll
  for (int r = 0; r < 8; ++r)
    out[(size_t)r * D] = c[r];
}

// ---------------------------------------------------------------------------
// One wave per (m, node):
//  lane = k : neighbor score = sum_d tanh(ps+po+b)*x ; softmax over K with
//  the reference's dense-baseline denominator; then lane = E/4-chunk:
//  H = src + sum_k A_k * other[idx_k]; fused semantic projection ->
//  atomicAdd into beta_accum[m] (pre-scaled by 1/N for the mean).
// ---------------------------------------------------------------------------
__global__ __launch_bounds__(256) void attn_kernel(
    const float* __restrict__ src, const float* __restrict__ other,
    const int* __restrict__ nbrs,                  // [M, N, K]
    const float* __restrict__ proj_src,            // [M, N, D]
    const float* __restrict__ proj_other,          // [M, N, D]
    const float* __restrict__ Xv,                  // [M, D]
    const float* __restrict__ Bp,                  // [M, D]
    const float* __restrict__ Wq,                  // [M, E, D]
    const float* __restrict__ Bq,                  // [M, D]
    const float* __restrict__ Qv,                  // [M, D]
    float* __restrict__ H,                         // [M, N, E]
    float* __restrict__ beta_accum)                // [M]
{
  __shared__ float sA[8 * K];
  __shared__ int   sI[8 * K];
  __shared__ float sH[8 * E];

  const int lane = threadIdx.x & 31;
  const int wid  = threadIdx.x >> 5;
  const int w    = blockIdx.x * 8 + wid;           // 0 .. M*N-1
  const int m    = w >> 13;                        // / N
  const int node = w & (N - 1);

  const int idx = nbrs[(size_t)w * K + lane];      // this lane's neighbor

  const float4* po = (const float4*)(proj_other + ((size_t)m * N + idx) * D);
  const float4* ps = (const float4*)(proj_src + (size_t)w * D);
  const float4* bp = (const float4*)(Bp + m * D);
  const float4* xv = (const float4*)(Xv + m * D);

  float s = 0.f;
#pragma unroll
  for (int j = 0; j < D / 4; ++j) {
    float4 a = po[j], p = ps[j], b = bp[j], x = xv[j];
    s += fast_tanh(p.x + a.x + b.x) * x.x;
    s += fast_tanh(p.y + a.y + b.y) * x.y;
    s += fast_tanh(p.z + a.z + b.z) * x.z;
    s += fast_tanh(p.w + a.w + b.w) * x.w;
  }

  // softmax over K with dense baseline (metapath 0: -1e-9, else 1/N)
  const float baseline = (m == 0) ? -1e-9f : (1.0f / (float)N);
  float mx = s;
#pragma unroll
  for (int off = 16; off >= 1; off >>= 1) mx = fmaxf(mx, __shfl_xor(mx, off, 32));
  mx = fmaxf(mx, baseline);
  const float ex = __expf(s - mx);
  float esum = ex;
#pragma unroll
  for (int off = 16; off >= 1; off >>= 1) esum += __shfl_xor(esum, off, 32);
  const float denom = esum + (float)(N - K) * __expf(baseline - mx);
  const float a_k = ex / denom;

  sA[wid * K + lane] = a_k;
  sI[wid * K + lane] = idx;
  __syncthreads();

  // H row: lane owns one float4 chunk of E=128
  float4 acc = ((const float4*)(src + (size_t)node * E))[lane];
  for (int k = 0; k < K; ++k) {
    const float wk = sA[wid * K + k];
    const int   jk = sI[wid * K + k];
    const float4 o = ((const float4*)(other + (size_t)jk * E))[lane];
    acc.x = fmaf(wk, o.x, acc.x);
    acc.y = fmaf(wk, o.y, acc.y);
    acc.z = fmaf(wk, o.z, acc.z);
    acc.w = fmaf(wk, o.w, acc.w);
  }
  ((float4*)(H + (size_t)w * E))[lane] = acc;
  ((float4*)(sH + wid * E))[lane] = acc;
  __syncthreads();

  // semantic attention partial: lane handles d = lane and lane + 32
  const float* hrow = sH + wid * E;
  float part = 0.f;
#pragma unroll
  for (int t = 0; t < 2; ++t) {
    const int d = lane + 32 * t;
    float ssum = Bq[m * D + d];
    const float* wq = Wq + (size_t)m * E * D + d;
    for (int e2 = 0; e2 < E; ++e2)
      ssum = fmaf(hrow[e2], wq[(size_t)e2 * D], ssum);
    part += fast_tanh(ssum) * Qv[m * D + d];
  }
#pragma unroll
  for (int off = 16; off >= 1; off >>= 1) part += __shfl_xor(part, off, 32);
  if (lane == 0) atomicAdd(&beta_accum[m], part * (1.0f / (float)N));
}

// ---------------------------------------------------------------------------
// out[n, e] = sum_m softmax(beta_raw)[m] * H[m, n, e]
// ---------------------------------------------------------------------------
__global__ __launch_bounds__(256) void combine_kernel(
    const float* __restrict__ H, const float* __restrict__ beta_raw,
    float* __restrict__ out)
{
  const int i = blockIdx.x * 256 + threadIdx.x;     // over N*E/4 float4s
  const float b0 = beta_raw[0], b1 = beta_raw[1], b2 = beta_raw[2], b3 = beta_raw[3];
  const float bm = fmaxf(fmaxf(b0, b1), fmaxf(b2, b3));
  float e0 = __expf(b0 - bm), e1 = __expf(b1 - bm);
  float e2 = __expf(b2 - bm), e3 = __expf(b3 - bm);
  const float inv = 1.0f / (e0 + e1 + e2 + e3);
  e0 *= inv; e1 *= inv; e2 *= inv; e3 *= inv;

  const float4* H4 = (const float4*)H;
  const size_t stride = (size_t)N * E / 4;
  const float4 h0 = H4[i], h1 = H4[i + stride];
  const float4 h2 = H4[i + 2 * stride], h3 = H4[i + 3 * stride];
  float4 r;
  r.x = e0 * h0.x + e1 * h1.x + e2 * h2.x + e3 * h3.x;
  r.y = e0 * h0.y + e1 * h1.y + e2 * h2.y + e3 * h3.y;
  r.z = e0 * h0.z + e1 * h1.z + e2 * h2.z + e3 * h3.z;
  r.w = e0 * h0.w + e1 * h1.w + e2 * h2.w + e3 * h3.w;
  ((float4*)out)[i] = r;
}

__global__ void zero_kernel(float* __restrict__ p, int n) {
  if ((int)threadIdx.x < n) p[threadIdx.x] = 0.f;
}

// ---------------------------------------------------------------------------
extern "C" void kernel_launch(void* const* d_in, const int* in_sizes, int n_in,
                              void* d_out, int out_size, void* d_ws, size_t ws_size,
                              hipStream_t stream) {
  (void)in_sizes; (void)n_in; (void)out_size; (void)ws_size;
  const float* user     = (const float*)d_in[0];
  const float* product  = (const float*)d_in[1];
  const float* V        = (const float*)d_in[2];   // [2,M,E,D]
  const float* X        = (const float*)d_in[3];   // [2,M,1,D]
  const float* W_p      = (const float*)d_in[4];   // [2,M,E,D]
  const float* B_p      = (const float*)d_in[5];   // [2,M,D]
  const float* W_q      = (const float*)d_in[6];   // [2,M,E,D]
  const float* B_q      = (const float*)d_in[7];   // [2,M,D]
  const float* Q        = (const float*)d_in[8];   // [2,M,1,D]
  const int* user_nbrs    = (const int*)d_in[9];   // [M,N,K]
  const int* product_nbrs = (const int*)d_in[10];  // [M,N,K]
  float* out = (float*)d_out;                      // [2, N, E] concatenated

  // workspace layout: proj_src | proj_other | H | beta  (~32 MiB total)
  float* proj_src   = (float*)d_ws;                         // M*N*D
  float* proj_other = proj_src   + (size_t)M * N * D;       // M*N*D
  float* Hbuf       = proj_other + (size_t)M * N * D;       // M*N*E
  float* beta       = Hbuf       + (size_t)M * N * E;       // M

  const size_t wED = (size_t)M * E * D;
  const size_t wD  = (size_t)M * D;

  for (int p = 0; p < 2; ++p) {
    const float* src   = (p == 0) ? user : product;
    // phase 1 attends over the UPDATED user embeddings (phase-0 output)
    const float* other = (p == 0) ? product : out;
    const int*   nbrs  = (p == 0) ? user_nbrs : product_nbrs;
    float* phase_out = out + (size_t)p * N * E;

    gemm_proj_kernel<<<(M * (N / 16) * (D / 16)) / 8, 256, 0, stream>>>(
        src, V + p * wED, proj_src);
    gemm_proj_kernel<<<(M * (N / 16) * (D / 16)) / 8, 256, 0, stream>>>(
        other, W_p + p * wED, proj_other);
    zero_kernel<<<1, 32, 0, stream>>>(beta, M);
    attn_kernel<<<(M * N) / 8, 256, 0, stream>>>(
        src, other, nbrs, proj_src, proj_other,
        X + p * wD, B_p + p * wD, W_q + p * wED, B_q + p * wD, Q + p * wD,
        Hbuf, beta);
    combine_kernel<<<(N * E / 4) / 256, 256, 0, stream>>>(Hbuf, beta, phase_out);
  }
}